// GCNBlock_5952824672505
// MI455X (gfx1250) — compile-verified
//
#include <hip/hip_runtime.h>
#include <hip/hip_bf16.h>

// ---------------------------------------------------------------------------
// GCN layer: out = relu(A_norm @ (X @ W) + b), A_norm given as edge list.
//   Stage 1: H = X @ W           (fp32 WMMA 16x16x4, LDS-staged tiles)
//   Stage 2: out += w_e * H[src] scattered to out[dst]  (wave-per-edge,
//            float4 gather + 4x global_atomic_add_f32 per lane)
//   Stage 3: out = relu(out + b) in place
// H (51.2 MB) lives in d_ws and fits in the 192 MB L2, so stage-2 gathers
// and atomics resolve in L2 rather than HBM.
// ---------------------------------------------------------------------------

typedef float v2f __attribute__((ext_vector_type(2)));
typedef float v8f __attribute__((ext_vector_type(8)));

#define D_FEAT 128
#define HIDDEN 128
#define LDA 132   // A-tile LDS row stride (floats): bank pairs 4m+k / 4m+k+2, conflict-free b64
#define SW  288   // W pair-row stride (floats): 288%64==32 -> lane halves use disjoint bank halves

// ---- Stage 1: H = X @ W via V_WMMA_F32_16X16X4_F32 -------------------------
// Block = 256 threads = 8 waves. Each wave computes a 16x128 slab of H.
// W is staged K-paired: sW2[k>>1][n] = (W[k][n], W[k+1][n]) so each B fragment
// is a single contiguous ds_load_b64 straight into an even-aligned VGPR pair.
__global__ __launch_bounds__(256) void gcn_gemm_wmma(
    const float* __restrict__ X, const float* __restrict__ W,
    float* __restrict__ H, int n_nodes)
{
    __shared__ float sW2[(D_FEAT / 2) * SW];   // 64 pair-rows * 288 * 4B = 73,728 B
    __shared__ float sA[8 * 16 * LDA];         // 8 waves * 16x128 tile      = 67,584 B

    const int tid  = threadIdx.x;
    const int wave = tid >> 5;
    const int lane = tid & 31;

    // Cooperative load of W (128x128 row-major) into K-paired LDS layout.
    for (int i = tid; i < (D_FEAT * HIDDEN) / 4; i += 256) {
        const int k  = i >> 5;               // 32 float4 per source row
        const int c4 = (i & 31) << 2;
        const float4 w4 = reinterpret_cast<const float4*>(W)[i];
        float* d = &sW2[(k >> 1) * SW + (c4 << 1) + (k & 1)];
        d[0] = w4.x; d[2] = w4.y; d[4] = w4.z; d[6] = w4.w;
    }

    const int mtile   = blockIdx.x * 8 + wave;
    const bool active = (mtile * 16) < n_nodes;   // N_NODES % 16 == 0, no partial tiles
    float* sa = &sA[wave * 16 * LDA];
    if (active) {
        const float* Xt = X + (size_t)mtile * 16 * D_FEAT;
        for (int i = lane; i < 16 * 32; i += 32) {   // 512 float4 per 16x128 tile
            const int row = i >> 5;
            const int c4  = (i & 31) << 2;
            const float4 x4 = reinterpret_cast<const float4*>(Xt)[i];
            float* d = &sa[row * LDA + c4];
            d[0] = x4.x; d[1] = x4.y; d[2] = x4.z; d[3] = x4.w;
        }
    }
    __syncthreads();
    if (!active) return;

    v8f acc[8] = {};                 // 8 N-tiles of 16x16 fp32 accumulators

    const int mrow = lane & 15;            // A: row within M-tile
    const int khi  = (lane >> 4) << 1;     // lane half selects K / K+2
    const int ncol = lane & 15;            // B/D: column within N-tile

    for (int k = 0; k < D_FEAT; k += 4) {
        // A fragment (16x4 fp32): VGPR0 = K0|K2 across lane halves, VGPR1 = K1|K3.
        // Contiguous + 8B-aligned -> single ds_load_b64.
        const v2f a = *reinterpret_cast<const v2f*>(&sa[mrow * LDA + k + khi]);
        const int prow = (k + khi) >> 1;   // even since k%4==0, khi in {0,2}
#pragma unroll
        for (int t = 0; t < 8; ++t) {
            // B fragment (4x16 fp32): rows K0|K2 striped over lanes; K-paired
            // layout makes (W[K][n], W[K+1][n]) one ds_load_b64.
            const v2f b = *reinterpret_cast<const v2f*>(
                &sW2[prow * SW + ((t * 16 + ncol) << 1)]);
            acc[t] = __builtin_amdgcn_wmma_f32_16x16x4_f32(
                false, a, false, b, (short)0, acc[t], false, false);
        }
    }

    // D layout: VGPR r holds rows r (lanes 0-15) and r+8 (lanes 16-31)
    float* Ht = H + (size_t)mtile * 16 * HIDDEN;
    const int rbase = (lane >> 4) << 3;
#pragma unroll
    for (int t = 0; t < 8; ++t)
#pragma unroll
        for (int r = 0; r < 8; ++r)
            Ht[(size_t)(r + rbase) * HIDDEN + t * 16 + ncol] = acc[t][r];
}

// ---- Stage 2: scatter-add messages, one wave (32 lanes) per edge -----------
__global__ __launch_bounds__(256) void gcn_scatter(
    const float* __restrict__ H, const long long* __restrict__ ei,
    const float* __restrict__ ew, float* __restrict__ out, int n_edges)
{
    const int wave = threadIdx.x >> 5;
    const int lane = threadIdx.x & 31;
    const int e = blockIdx.x * 8 + wave;
    if (e >= n_edges) return;

    const long long src = ei[e];
    const long long dst = ei[(size_t)n_edges + e];
    const float w = ew[e];

    // 128 floats per edge: lane owns 4 contiguous features (float4 gather)
    const float4 m = reinterpret_cast<const float4*>(H + (size_t)src * HIDDEN)[lane];
    float* o = out + (size_t)dst * HIDDEN + lane * 4;
    unsafeAtomicAdd(o + 0, m.x * w);   // -> global_atomic_add_f32 (no CAS loop)
    unsafeAtomicAdd(o + 1, m.y * w);
    unsafeAtomicAdd(o + 2, m.z * w);
    unsafeAtomicAdd(o + 3, m.w * w);
}

// ---- Stage 3: out = relu(out + b), in place, float4 wide -------------------
__global__ __launch_bounds__(256) void gcn_bias_relu(
    float* __restrict__ out, const float* __restrict__ b, int n4)
{
    const int i = blockIdx.x * 256 + threadIdx.x;
    if (i >= n4) return;
    float4 v = reinterpret_cast<float4*>(out)[i];
    const float4 bb = reinterpret_cast<const float4*>(b)[i & 31]; // HIDDEN/4 == 32
    v.x = fmaxf(v.x + bb.x, 0.0f);
    v.y = fmaxf(v.y + bb.y, 0.0f);
    v.z = fmaxf(v.z + bb.z, 0.0f);
    v.w = fmaxf(v.w + bb.w, 0.0f);
    reinterpret_cast<float4*>(out)[i] = v;
}

extern "C" void kernel_launch(void* const* d_in, const int* in_sizes, int n_in,
                              void* d_out, int out_size, void* d_ws, size_t ws_size,
                              hipStream_t stream) {
    const float*     X  = (const float*)d_in[0];       // [N, 128]
    const long long* ei = (const long long*)d_in[1];   // [2, E] int64
    const float*     ew = (const float*)d_in[2];       // [E]
    const float*     W  = (const float*)d_in[3];       // [128, 128]
    const float*     b  = (const float*)d_in[4];       // [128]
    float* out = (float*)d_out;                        // [N, 128]
    float* H   = (float*)d_ws;                         // [N, 128] scratch (51.2 MB)

    const int n_nodes = in_sizes[0] / D_FEAT;
    const int n_edges = in_sizes[2];

    // Zero the accumulation target every launch (graph-capture safe).
    hipMemsetAsync(d_out, 0, (size_t)out_size * sizeof(float), stream);

    const int mtiles = (n_nodes + 15) / 16;
    gcn_gemm_wmma<<<(mtiles + 7) / 8, 256, 0, stream>>>(X, W, H, n_nodes);

    gcn_scatter<<<(n_edges + 7) / 8, 256, 0, stream>>>(H, ei, ew, out, n_edges);

    const int n4 = (n_nodes * HIDDEN) / 4;
    gcn_bias_relu<<<(n4 + 255) / 256, 256, 0, stream>>>(out, b, n4);
}